// LightGCN_12893491823093
// MI455X (gfx1250) — compile-verified
//
#include <hip/hip_runtime.h>

// ---------------------------------------------------------------------------
// LightGCN on MI455X (gfx1250, wave32).
//
// Key reduction: propagation is linear & identical per feature dim, so the
// per-node feature ROW-SUM propagates as a scalar:
//   s0[n]   = sum_d emb[node_indices[n], d]
//   s_l[n]  = in_norm[n] * sum_{e:dst=n} out_norm[src_e] * s_{l-1}[src_e]
//   total[n]= (s0+s1+s2+s3)[n] / 256
// Output: total gathered at user/pos/neg ids -> [3, B].
//
// The 128MB row-sum pass uses V_WMMA_F32_16X16X4_F32 (emb_tile(16x64) x ones),
// staged through LDS with coalesced global loads; full f32 precision.
// ---------------------------------------------------------------------------

typedef float v2f __attribute__((ext_vector_type(2)));
typedef float v8f __attribute__((ext_vector_type(8)));

#if defined(__has_builtin)
#  if __has_builtin(__builtin_amdgcn_wmma_f32_16x16x4_f32)
#    define HAVE_WMMA_F32X4 1
#  endif
#endif
#ifndef HAVE_WMMA_F32X4
#  define HAVE_WMMA_F32X4 0
#endif

#define LG_WAVES_PER_BLOCK 4
#define LG_EMB_DIM 64
#define LG_LDS_PITCH 68   // 64 + 4 pad floats: 4-bank shift per row, 8B aligned

__global__ void lg_zero3(float* __restrict__ a, float* __restrict__ b,
                         float* __restrict__ c, int n) {
  int i = blockIdx.x * blockDim.x + threadIdx.x;
  if (i < n) { a[i] = 0.f; b[i] = 0.f; c[i] = 0.f; }
}

__global__ void lg_degree(const int* __restrict__ src, const int* __restrict__ dst,
                          float* __restrict__ degO, float* __restrict__ degI, int E) {
  int e = blockIdx.x * blockDim.x + threadIdx.x;
  if (e < E) {
    atomicAdd(&degO[src[e]], 1.0f);
    atomicAdd(&degI[dst[e]], 1.0f);
  }
}

// Row-sum of 16 embedding rows per wave via V_WMMA_F32_16X16X4_F32.
// s[n] = tot[n] = sum_d emb[nidx[n]*64 + d]
__global__ void __launch_bounds__(32 * LG_WAVES_PER_BLOCK)
lg_rowsum_wmma(const float* __restrict__ emb, const int* __restrict__ nidx,
               float* __restrict__ s, float* __restrict__ tot, int nTiles) {
  __shared__ float smem[LG_WAVES_PER_BLOCK][16 * LG_LDS_PITCH];
  const int lane = threadIdx.x & 31;
  const int wid  = threadIdx.x >> 5;
  const int t    = blockIdx.x * LG_WAVES_PER_BLOCK + wid;   // wave-uniform
  if (t >= nTiles) return;                                  // whole wave exits
  const int base = t * 16;

  int myIdx = 0;
  if (lane < 16) myIdx = nidx[base + lane];

  // Stage the 16x64 tile into LDS, fully coalesced: each iteration the wave
  // loads two complete 256B rows (float4 per lane).
  const float4* emb4 = (const float4*)emb;
  const int q    = lane & 15;
  const int half = lane >> 4;
#pragma unroll
  for (int j = 0; j < 8; ++j) {
    const int row  = 2 * j + half;
    const int ridx = __shfl(myIdx, row, 32);
    float4 v = emb4[(size_t)ridx * 16 + q];
    float* dp = &smem[wid][row * LG_LDS_PITCH + q * 4];
    ((float2*)dp)[0] = make_float2(v.x, v.y);
    ((float2*)dp)[1] = make_float2(v.z, v.w);
  }
  // Same-wave LDS ops are in-order (DScnt); each wave owns its LDS slab,
  // so no cross-wave barrier is needed.

#if HAVE_WMMA_F32X4
  // A fragment (16x4 f32): lane L holds M = L%16; lanes 0-15 -> K = kb,kb+1;
  // lanes 16-31 -> K = kb+2,kb+3.  B = ones(4x16) => D[m][*] = rowsum(A_m).
  v8f c = {};
  const v2f ones = {1.0f, 1.0f};
  const int row  = lane & 15;
  const int koff = half * 2;
#pragma unroll
  for (int kk = 0; kk < 16; ++kk) {
    const float2 af = *(const float2*)&smem[wid][row * LG_LDS_PITCH + kk * 4 + koff];
    v2f a; a.x = af.x; a.y = af.y;
    c = __builtin_amdgcn_wmma_f32_16x16x4_f32(false, a, false, ones,
                                              (short)0, c, false, false);
  }
  // D layout: VGPR r, lanes 0-15 -> M=r; lanes 16-31 -> M=8+r (any column N).
  if (lane == 0) {
#pragma unroll
    for (int r = 0; r < 8; ++r) { s[base + r] = c[r]; tot[base + r] = c[r]; }
  } else if (lane == 16) {
#pragma unroll
    for (int r = 0; r < 8; ++r) { s[base + 8 + r] = c[r]; tot[base + 8 + r] = c[r]; }
  }
#else
  // Fallback (also taken by the host-side parse where the builtin is absent).
  if (lane < 16) {
    float acc = 0.f;
    for (int d = 0; d < LG_EMB_DIM; ++d) acc += smem[wid][lane * LG_LDS_PITCH + d];
    s[base + lane] = acc;
    tot[base + lane] = acc;
  }
#endif
}

__global__ void lg_rowsum_tail(const float* __restrict__ emb,
                               const int* __restrict__ nidx,
                               float* __restrict__ s, float* __restrict__ tot,
                               int start, int N) {
  int i = start + blockIdx.x * blockDim.x + threadIdx.x;
  if (i < N) {
    const float* r = emb + (size_t)nidx[i] * LG_EMB_DIM;
    float acc = 0.f;
    for (int d = 0; d < LG_EMB_DIM; ++d) acc += r[d];
    s[i] = acc;
    tot[i] = acc;
  }
}

// In-place: onorm/inorm buffers arrive holding the (float) degrees.
__global__ void lg_norm_prescale(float* onorm, float* inorm,
                                 const float* __restrict__ s, float* __restrict__ p,
                                 int n) {
  int i = blockIdx.x * blockDim.x + threadIdx.x;
  if (i < n) {
    float o  = 1.0f / sqrtf(fmaxf(onorm[i], 1.0f));
    float in_ = 1.0f / sqrtf(fmaxf(inorm[i], 1.0f));
    onorm[i] = o;
    inorm[i] = in_;
    p[i] = o * s[i];
  }
}

__global__ void lg_scatter(const int* __restrict__ src, const int* __restrict__ dst,
                           const float* __restrict__ p, float* __restrict__ agg, int E) {
  int e = blockIdx.x * blockDim.x + threadIdx.x;
  if (e < E) atomicAdd(&agg[dst[e]], p[src[e]]);
}

__global__ void lg_finish(float* __restrict__ agg, const float* __restrict__ inorm,
                          const float* __restrict__ onorm, float* __restrict__ tot,
                          float* __restrict__ p, int n) {
  int i = blockIdx.x * blockDim.x + threadIdx.x;
  if (i < n) {
    float snew = inorm[i] * agg[i];
    tot[i] += snew;
    p[i]   = onorm[i] * snew;
    agg[i] = 0.f;                // ready for next layer's scatter
  }
}

__global__ void lg_gather(const float* __restrict__ tot, const int* __restrict__ uid,
                          const int* __restrict__ pid, const int* __restrict__ nid,
                          float* __restrict__ out, int B, float invM) {
  int i = blockIdx.x * blockDim.x + threadIdx.x;
  if (i < B) {
    out[i]         = tot[uid[i]] * invM;
    out[B + i]     = tot[pid[i]] * invM;
    out[2 * B + i] = tot[nid[i]] * invM;
  }
}

extern "C" void kernel_launch(void* const* d_in, const int* in_sizes, int n_in,
                              void* d_out, int out_size, void* d_ws, size_t ws_size,
                              hipStream_t stream) {
  const float* emb  = (const float*)d_in[0];
  const int*   nidx = (const int*)d_in[1];
  const int*   src  = (const int*)d_in[2];
  const int*   dst  = (const int*)d_in[3];
  const int*   uid  = (const int*)d_in[4];
  const int*   pid  = (const int*)d_in[5];
  const int*   nid  = (const int*)d_in[6];
  float*       out  = (float*)d_out;

  const int N = in_sizes[1];           // 500,000 nodes
  const int E = in_sizes[2];           // 1,250,000 edges
  const int B = in_sizes[4];           // 4096
  const int D = in_sizes[0] / N;       // 64
  const int NUM_LAYERS = 3;
  const float invM = 1.0f / (float)((NUM_LAYERS + 1) * D);   // mean over concat dim

  // Workspace layout: 6 arrays of N floats (12 MB)
  float* W     = (float*)d_ws;
  float* s     = W;                    // current layer row-sums (s0 at first)
  float* tot   = W + (size_t)N;        // running total of row-sums
  float* p     = W + 2 * (size_t)N;    // out_norm-prescaled messages
  float* agg   = W + 3 * (size_t)N;    // scatter accumulator
  float* onorm = W + 4 * (size_t)N;    // degrees -> out_norm (in place)
  float* inorm = W + 5 * (size_t)N;    // degrees -> in_norm  (in place)

  const int TPB = 256;
  const int gN  = (N + TPB - 1) / TPB;
  const int gE  = (E + TPB - 1) / TPB;
  const int gB  = (B + TPB - 1) / TPB;

  // 1) zero accumulators (ws is poisoned, and agg must start at 0 every call)
  lg_zero3<<<gN, TPB, 0, stream>>>(agg, onorm, inorm, N);

  // 2) degrees via float atomics
  lg_degree<<<gE, TPB, 0, stream>>>(src, dst, onorm, inorm, E);

  // 3) row-sums of gathered embeddings (WMMA f32 16x16x4 path)
  const int nTiles = N / 16;
  if (nTiles > 0) {
    const int gT = (nTiles + LG_WAVES_PER_BLOCK - 1) / LG_WAVES_PER_BLOCK;
    lg_rowsum_wmma<<<gT, 32 * LG_WAVES_PER_BLOCK, 0, stream>>>(emb, nidx, s, tot, nTiles);
  }
  const int tailStart = nTiles * 16;
  if (tailStart < N) {
    lg_rowsum_tail<<<1, TPB, 0, stream>>>(emb, nidx, s, tot, tailStart, N);
  }

  // 4) norms + prescale p = out_norm * s0
  lg_norm_prescale<<<gN, TPB, 0, stream>>>(onorm, inorm, s, p, N);

  // 5) three propagation layers on the scalar row-sums
  for (int l = 0; l < NUM_LAYERS; ++l) {
    lg_scatter<<<gE, TPB, 0, stream>>>(src, dst, p, agg, E);
    lg_finish<<<gN, TPB, 0, stream>>>(agg, inorm, onorm, tot, p, N);
  }

  // 6) gather user/pos/neg -> [3, B], apply mean divisor
  lg_gather<<<gB, TPB, 0, stream>>>(tot, uid, pid, nid, out, B, invM);
}